// ProbSparseAttention_25958782337325
// MI455X (gfx1250) — compile-verified
//
#include <hip/hip_runtime.h>
#include <math.h>

// ProbSparse attention for MI455X (gfx1250), fp32 throughout.
// B=4, H=16, T=4096, Dh=64, top_u=134 (padded to 144 = 9 tiles of 16).
// Round 2: double-buffered K/V staging via GLOBAL_LOAD_ASYNC_TO_LDS_B128
// (ASYNCcnt-tracked) + 32-keys-per-iteration flash loop.

typedef float v2f __attribute__((ext_vector_type(2)));
typedef float v8f __attribute__((ext_vector_type(8)));

#define WMMA_F32_4(a, b, c)                                                   \
  __builtin_amdgcn_wmma_f32_16x16x4_f32(false, (a), false, (b), (short)0,     \
                                        (c), false, false)

static constexpr int Bdim = 4;
static constexpr int Hdim = 16;
static constexpr int Tdim = 4096;
static constexpr int Dh   = 64;
static constexpr int TOPU = 134;
static constexpr int UPAD = 144;      // 9 tiles of 16
static constexpr int NBH  = Bdim * Hdim;
static constexpr int LDK  = 68;       // padded LDS row stride (floats)

// CDNA5 async global->LDS copy (no VGPR round-trip, tracked by ASYNCcnt).
// LDS dest = low 32 bits of the flat shared-memory pointer (aperture rule).
static __device__ __forceinline__ void async_ld_b128(unsigned lds_byte,
                                                     const float* g) {
  asm volatile("global_load_async_to_lds_b128 %0, %1, off"
               :: "v"(lds_byte), "v"(g)
               : "memory");
}

static __device__ __forceinline__ void wait_asynccnt0() {
#if __has_builtin(__builtin_amdgcn_s_wait_asynccnt)
  __builtin_amdgcn_s_wait_asynccnt(0);
#else
  asm volatile("s_wait_asynccnt 0x0" ::: "memory");
#endif
}

// ---------------------------------------------------------------------------
// 1) q_score[bh*T + t] = sum_d Q[bh,t,d]^2   (16 lanes per row, shfl reduce)
__global__ __launch_bounds__(256) void qscore_kernel(const float* __restrict__ Q,
                                                     float* __restrict__ score) {
  int gl = blockIdx.x * 256 + threadIdx.x;   // 16 lanes per query row
  int q = gl >> 4;
  int part = gl & 15;
  const float4 v = *(const float4*)(Q + (size_t)q * Dh + part * 4);
  float d = v.x * v.x + v.y * v.y + v.z * v.z + v.w * v.w;
  d += __shfl_xor(d, 1);
  d += __shfl_xor(d, 2);
  d += __shfl_xor(d, 4);
  d += __shfl_xor(d, 8);
  if (part == 0) score[q] = d;
}

// ---------------------------------------------------------------------------
// 2a) partial V sums: 8 slices of 512 keys per (b,h)
__global__ __launch_bounds__(256) void vsum_part(const float* __restrict__ V,
                                                 float* __restrict__ vpart) {
  int blk = blockIdx.x;                 // 0..511
  int bh = blk >> 3, slice = blk & 7;
  int d = threadIdx.x & 63, g = threadIdx.x >> 6;
  const float* vp = V + ((size_t)bh * Tdim + slice * 512) * Dh;
  float acc = 0.f;
  for (int k = g; k < 512; k += 4) acc += vp[(size_t)k * Dh + d];
  __shared__ float red[256];
  red[threadIdx.x] = acc;
  __syncthreads();
  if (threadIdx.x < 64) {
    vpart[blk * 64 + threadIdx.x] = red[threadIdx.x] + red[threadIdx.x + 64] +
                                    red[threadIdx.x + 128] +
                                    red[threadIdx.x + 192];
  }
}

// 2b) deterministic final reduce: vmean[bh*64+d] = mean over 4096 keys
__global__ __launch_bounds__(256) void vmean_reduce(const float* __restrict__ vpart,
                                                    float* __restrict__ vmean) {
  int gid = blockIdx.x * 256 + threadIdx.x;   // 0..4095
  int bh = gid >> 6, d = gid & 63;
  float s = 0.f;
#pragma unroll
  for (int sl = 0; sl < 8; ++sl) s += vpart[(bh * 8 + sl) * 64 + d];
  vmean[gid] = s * (1.0f / (float)Tdim);
}

// ---------------------------------------------------------------------------
// 3) top-134 per (b,h) via radix-select on float bits (scores >= 0 so the
//    IEEE bit pattern is monotonic as uint). 32 bit-descent passes over LDS.
__global__ __launch_bounds__(256) void topk_kernel(const float* __restrict__ score,
                                                   int* __restrict__ idxbuf) {
  const int bh = blockIdx.x;
  const int tid = threadIdx.x;
  __shared__ unsigned int sv[Tdim];
  __shared__ int red[8];
  __shared__ int bcast;
  __shared__ int cnt_gt, cnt_eq;

  const float* s = score + (size_t)bh * Tdim;
  for (int j = tid; j < Tdim; j += 256) sv[j] = __float_as_uint(s[j]);
  if (tid == 0) { cnt_gt = 0; cnt_eq = 0; }
  __syncthreads();

  unsigned int prefix = 0;
  int need = TOPU;
  for (int b = 31; b >= 0; --b) {
    unsigned int pat = (prefix >> b) | 1u;
    int c = 0;
    for (int j = tid; j < Tdim; j += 256) c += (int)((sv[j] >> b) == pat);
    c += __shfl_xor(c, 1);
    c += __shfl_xor(c, 2);
    c += __shfl_xor(c, 4);
    c += __shfl_xor(c, 8);
    c += __shfl_xor(c, 16);
    if ((tid & 31) == 0) red[tid >> 5] = c;
    __syncthreads();
    if (tid == 0) {
      int t = 0;
#pragma unroll
      for (int w = 0; w < 8; ++w) t += red[w];
      bcast = t;
    }
    __syncthreads();
    int total = bcast;
    if (total >= need) prefix |= (1u << b);
    else need -= total;
    __syncthreads();
  }
  // prefix == value of the 134th largest; need == #equal-values to keep.
  for (int j = tid; j < Tdim; j += 256) {
    if (sv[j] > prefix) {
      int pos = atomicAdd(&cnt_gt, 1);
      idxbuf[bh * UPAD + pos] = j;
    }
  }
  __syncthreads();
  int base = cnt_gt;
  for (int j = tid; j < Tdim; j += 256) {
    if (sv[j] == prefix) {
      int pos = atomicAdd(&cnt_eq, 1);
      if (base + pos < TOPU) idxbuf[bh * UPAD + base + pos] = j;
    }
  }
  if (tid < UPAD - TOPU) idxbuf[bh * UPAD + TOPU + tid] = 0;  // valid pad rows
}

// ---------------------------------------------------------------------------
// 4) out[bh,t,:] = vmean[bh,:]  (float4 coalesced broadcast)
__global__ __launch_bounds__(256) void fill_kernel(const float* __restrict__ vmean,
                                                   float* __restrict__ out) {
  size_t f = (size_t)blockIdx.x * 256 + threadIdx.x;  // float4 index
  int d4 = (int)(f & 15);
  int bh = (int)(f >> 16);                            // 65536 float4 per bh
  float4 v = *(const float4*)(vmean + bh * 64 + d4 * 4);
  *(float4*)(out + f * 4) = v;
}

// ---------------------------------------------------------------------------
// 5) Flash attention for the 134 selected queries per (b,h).
//    One block per (b,h): 9 waves, one 16-query WMMA tile each.
//    32 keys per iteration, double-buffered async K/V staging.
__global__ __launch_bounds__(288) void attn_kernel(const float* __restrict__ Q,
                                                   const float* __restrict__ K,
                                                   const float* __restrict__ V,
                                                   const int* __restrict__ idxbuf,
                                                   float* __restrict__ out) {
  const int bh = blockIdx.x;              // 0..63
  const int wave = threadIdx.x >> 5;      // 0..8 == query tile
  const int lane = threadIdx.x & 31;
  const int half = lane >> 4;             // 0|1
  const int n16 = lane & 15;

  // Double-buffered 32-key K/V tiles, padded stride 68 floats/row.
  __shared__ float lds_k[2][32 * LDK];
  __shared__ float lds_v[2][32 * LDK];
  __shared__ float lds_p[9][16 * 18];     // per-wave P reshape buffer

  const int qt = wave;
  const float scale = 0.125f;             // 1/sqrt(64)

  const float* Kbh = K + (size_t)bh * Tdim * Dh;
  const float* Vbh = V + (size_t)bh * Tdim * Dh;

  // ---- loader thread setup (waves 0-7): each thread owns rows r0 and r0+16
  const bool loader = (threadIdx.x < 256);
  const int r0 = threadIdx.x >> 4;               // 0..15
  const int c4 = (threadIdx.x & 15) * 4;         // 0..60
  const unsigned row16b = 16 * LDK * 4;          // +16 rows in bytes
  unsigned lk[2], lv[2];
  lk[0] = (unsigned)(size_t)&lds_k[0][r0 * LDK + c4];
  lk[1] = (unsigned)(size_t)&lds_k[1][r0 * LDK + c4];
  lv[0] = (unsigned)(size_t)&lds_v[0][r0 * LDK + c4];
  lv[1] = (unsigned)(size_t)&lds_v[1][r0 * LDK + c4];
  const float* gk = Kbh + (size_t)r0 * Dh + c4;  // block-0 position
  const float* gv = Vbh + (size_t)r0 * Dh + c4;

  // ---- prologue: async-load key block 0 into buffer 0
  if (loader) {
    async_ld_b128(lk[0], gk);
    async_ld_b128(lk[0] + row16b, gk + 16 * Dh);
    async_ld_b128(lv[0], gv);
    async_ld_b128(lv[0] + row16b, gv + 16 * Dh);
    gk += 32 * Dh;
    gv += 32 * Dh;
  }

  // ---- per-wave Q fragments (A-layout: lane m=n16 holds K-pair {2h,2h+1})
  const int qrow = idxbuf[bh * UPAD + qt * 16 + n16];
  const float* qptr = Q + ((size_t)bh * Tdim + qrow) * Dh;
  v2f qa[16];
#pragma unroll
  for (int kc = 0; kc < 16; ++kc) {
    const float* p = qptr + 4 * kc + 2 * half;
    v2f t;
    t.x = p[0] * scale;
    t.y = p[1] * scale;
    qa[kc] = t;
  }

  v8f O0 = {}, O1 = {}, O2 = {}, O3 = {};
  float mrow[8], lrow[8];
#pragma unroll
  for (int g = 0; g < 8; ++g) {
    mrow[g] = -INFINITY;
    lrow[g] = 0.f;
  }
  float* pw = &lds_p[wave][0];

  wait_asynccnt0();
  __syncthreads();

  const int NIT = Tdim / 32;              // 128 iterations of 32 keys
  for (int it = 0; it < NIT; ++it) {
    const int cur = it & 1;
    // kick off async loads for the next 32-key block into the other buffer
    if (loader && (it + 1 < NIT)) {
      const int nxt = cur ^ 1;
      async_ld_b128(lk[nxt], gk);
      async_ld_b128(lk[nxt] + row16b, gk + 16 * Dh);
      async_ld_b128(lv[nxt], gv);
      async_ld_b128(lv[nxt] + row16b, gv + 16 * Dh);
      gk += 32 * Dh;
      gv += 32 * Dh;
    }

    const float* kbuf = &lds_k[cur][0];
    const float* vbuf = &lds_v[cur][0];

    // S = (Q*scale) @ K^T for the two 16-key subtiles (16 WMMAs each).
    v8f Sa = {}, Sb = {};
#pragma unroll
    for (int kc = 0; kc < 16; ++kc) {
      v2f ba = *(const v2f*)&kbuf[n16 * LDK + 4 * kc + 2 * half];
      v2f bb = *(const v2f*)&kbuf[16 * LDK + n16 * LDK + 4 * kc + 2 * half];
      Sa = WMMA_F32_4(qa[kc], ba, Sa);
      Sb = WMMA_F32_4(qa[kc], bb, Sb);
    }

    // Online softmax over 32 keys at once (one reduction pass for both tiles).
    // C/D layout: S[g] holds row (g + 8*half), col n16.
    float Pa[8], Pb[8], alpha[8];
#pragma unroll
    for (int g = 0; g < 8; ++g) {
      float t = fmaxf(Sa[g], Sb[g]);
      t = fmaxf(t, __shfl_xor(t, 1));
      t = fmaxf(t, __shfl_xor(t, 2));
      t = fmaxf(t, __shfl_xor(t, 4));
      t = fmaxf(t, __shfl_xor(t, 8));
      float mnew = fmaxf(mrow[g], t);
      float a = __expf(mrow[g] - mnew);
      float pa = __expf(Sa[g] - mnew);
      float pb = __expf(Sb[g] - mnew);
      float s = pa + pb;
      s += __shfl_xor(s, 1);
      s += __shfl_xor(s, 2);
      s += __shfl_xor(s, 4);
      s += __shfl_xor(s, 8);
      lrow[g] = lrow[g] * a + s;
      mrow[g] = mnew;
      alpha[g] = a;
      Pa[g] = pa;
      Pb[g] = pb;
    }
#pragma unroll
    for (int g = 0; g < 8; ++g) {
      O0[g] *= alpha[g];
      O1[g] *= alpha[g];
      O2[g] *= alpha[g];
      O3[g] *= alpha[g];
    }

    // O += P @ V for each subtile: reshape P through wave-private LDS
    // (LDS ops are in-order within a wave; no barrier needed).
#pragma unroll
    for (int sub = 0; sub < 2; ++sub) {
      const float* vsub = vbuf + sub * 16 * LDK;
      const float* Ps = sub ? Pb : Pa;
#pragma unroll
      for (int g = 0; g < 8; ++g) pw[(g + 8 * half) * 18 + n16] = Ps[g];
#pragma unroll
      for (int kc2 = 0; kc2 < 4; ++kc2) {
        v2f a = *(const v2f*)&pw[n16 * 18 + 4 * kc2 + 2 * half];
        int k0 = 4 * kc2 + 2 * half;
        v2f b0, b1, b2, b3;
        b0.x = vsub[k0 * LDK + 0 + n16];
        b0.y = vsub[(k0 + 1) * LDK + 0 + n16];
        b1.x = vsub[k0 * LDK + 16 + n16];
        b1.y = vsub[(k0 + 1) * LDK + 16 + n16];
        b2.x = vsub[k0 * LDK + 32 + n16];
        b2.y = vsub[(k0 + 1) * LDK + 32 + n16];
        b3.x = vsub[k0 * LDK + 48 + n16];
        b3.y = vsub[(k0 + 1) * LDK + 48 + n16];
        O0 = WMMA_F32_4(a, b0, O0);
        O1 = WMMA_F32_4(a, b1, O1);
        O2 = WMMA_F32_4(a, b2, O2);
        O3 = WMMA_F32_4(a, b3, O3);
      }
    }

    // Drain next-buffer async loads (loader waves), then one barrier:
    // everyone done reading buf[cur]; buf[cur^1] now visible to all waves.
    wait_asynccnt0();
    __syncthreads();
  }

  // Epilogue: scale by 1/l and scatter to selected query rows.
#pragma unroll
  for (int g = 0; g < 8; ++g) {
    int m = g + 8 * half;                  // row within tile
    int qpos = qt * 16 + m;
    if (qpos < TOPU) {
      int qi = idxbuf[bh * UPAD + qpos];
      float inv = 1.0f / lrow[g];
      float* op = out + ((size_t)bh * Tdim + qi) * Dh;
      op[0 + n16] = O0[g] * inv;
      op[16 + n16] = O1[g] * inv;
      op[32 + n16] = O2[g] * inv;
      op[48 + n16] = O3[g] * inv;
    }
  }
}

// ---------------------------------------------------------------------------
extern "C" void kernel_launch(void* const* d_in, const int* in_sizes, int n_in,
                              void* d_out, int out_size, void* d_ws,
                              size_t ws_size, hipStream_t stream) {
  (void)in_sizes; (void)n_in; (void)out_size; (void)ws_size;
  const float* Q = (const float*)d_in[0];
  const float* K = (const float*)d_in[1];
  const float* V = (const float*)d_in[2];
  float* out = (float*)d_out;

  char* ws = (char*)d_ws;
  float* score  = (float*)(ws);                               // 1 MB
  float* vpart  = (float*)(ws + (1u << 20));                  // 128 KB
  float* vmean  = (float*)(ws + (1u << 20) + (128u << 10));   // 16 KB
  int*   idxbuf = (int*)  (ws + (1u << 20) + (144u << 10));   // 36 KB

  qscore_kernel<<<(NBH * Tdim * 16) / 256, 256, 0, stream>>>(Q, score);
  vsum_part<<<NBH * 8, 256, 0, stream>>>(V, vpart);
  vmean_reduce<<<16, 256, 0, stream>>>(vpart, vmean);
  topk_kernel<<<NBH, 256, 0, stream>>>(score, idxbuf);
  fill_kernel<<<(NBH * Tdim * Dh / 4) / 256, 256, 0, stream>>>(vmean, out);
  attn_kernel<<<NBH, 288, 0, stream>>>(Q, K, V, idxbuf, out);
}